// RNNCell_78383153152162
// MI455X (gfx1250) — compile-verified
//
#include <hip/hip_runtime.h>

// Problem constants (match reference)
#define B_ROWS 4096
#define T_LEN  8192

// Tiling: one wave (32 lanes) per row; each lane owns a contiguous chunk of KCH
#define KCH        16                  // elements per lane chunk
#define STRD       20                  // padded dwords per chunk in LDS (16 data + 4 pad)
#define TILE_ELEMS 512                 // 32 * KCH
#define TILE_DW    (32 * STRD)         // 640 padded dwords per tile per array
#define NTILES     (T_LEN / TILE_ELEMS)// 16
#define WPB        4                   // waves per block (128 threads)
#define NBUF       3                   // triple buffering (safe reuse gap)

typedef float v4f __attribute__((ext_vector_type(4)));

// Low 32 bits of a generic pointer into __shared__ = workgroup-relative LDS byte
// address (generic LDS aperture keeps the LDS offset in addr[31:0]).
__device__ __forceinline__ unsigned lds_byte_off(const void* p) {
    return (unsigned)(size_t)p;
}

// Issue 4x GLOBAL_LOAD_ASYNC_TO_LDS_B128 covering 512 floats of one array.
// Global side is perfectly linear (lane*16B, +512B per instruction).
// LDS side scatters into the padded chunk layout so later per-chunk
// ds_load_b128 reads are (almost) conflict-free.
__device__ __forceinline__ void async_load_tile(const float* gsrc, unsigned ldsByte, int lane) {
#pragma unroll
    for (int j = 0; j < 4; ++j) {
        unsigned long long ga = (unsigned long long)(gsrc + j * 128 + lane * 4);
        // element g = 128j + 4*lane -> chunk c = 8j + lane/4, r = (4*lane)&12
        unsigned ldst = ldsByte + ((((unsigned)(8 * j + (lane >> 2)) * STRD) + ((lane * 4) & 12)) << 2);
        asm volatile("global_load_async_to_lds_b128 %0, %1, off"
                     :: "v"(ldst), "v"(ga)
                     : "memory");
    }
}

__global__ __launch_bounds__(WPB * 32) void visco_scan_kernel(
    const float* __restrict__ eps, const float* __restrict__ hs, float* __restrict__ out) {
    // [wave][buffer][array(eps,hs)][TILE_DW]
    __shared__ float smem[WPB * NBUF * 2 * TILE_DW];

    const int lane = threadIdx.x & 31;
    const int wave = threadIdx.x >> 5;
    const int row  = blockIdx.x * WPB + wave;
    if (row >= B_ROWS) return;   // uniform per wave

    float* wbase = &smem[wave * (NBUF * 2 * TILE_DW)];
    const float* rowE = eps + (size_t)row * T_LEN;
    const float* rowH = hs  + (size_t)row * T_LEN;
    float*       rowO = out + (size_t)row * T_LEN;

    // Prefetch tile 0 into buffer 0
    async_load_tile(rowE, lds_byte_off(wbase), lane);
    async_load_tile(rowH, lds_byte_off(wbase + TILE_DW), lane);

    float gTile    = 0.0f;  // gamma entering current tile
    float epsCarry = 0.0f;  // eps[t-1] for the first element of current tile

    int buf = 0;
    for (int tile = 0; tile < NTILES; ++tile) {
        int nb = buf + 1; if (nb == NBUF) nb = 0;

        if (tile + 1 < NTILES) {
            float* nbb = wbase + nb * (2 * TILE_DW);
            async_load_tile(rowE + (tile + 1) * TILE_ELEMS, lds_byte_off(nbb), lane);
            async_load_tile(rowH + (tile + 1) * TILE_ELEMS, lds_byte_off(nbb + TILE_DW), lane);
            // 8 async loads of tile t+1 may stay in flight; tile t's 8 are done
            asm volatile("s_wait_asynccnt 8" ::: "memory");
        } else {
            asm volatile("s_wait_asynccnt 0" ::: "memory");
        }

        const float* sE = wbase + buf * (2 * TILE_DW);
        const float* sH = sE + TILE_DW;

        // Read this lane's contiguous chunk from padded LDS (ds_load_b128 x4 per array)
        float e[KCH], h2[KCH];
        {
            const v4f* pe = (const v4f*)(sE + lane * STRD);
            const v4f* ph = (const v4f*)(sH + lane * STRD);
#pragma unroll
            for (int q = 0; q < 4; ++q) {
                v4f v = pe[q];
                e[4 * q + 0] = v.x; e[4 * q + 1] = v.y; e[4 * q + 2] = v.z; e[4 * q + 3] = v.w;
                v4f w = ph[q];
                h2[4 * q + 0] = w.x + w.x; h2[4 * q + 1] = w.y + w.y;
                h2[4 * q + 2] = w.z + w.z; h2[4 * q + 3] = w.w + w.w;  // 2h = h*(E/ETA)
            }
        }
        // eps_{t-1} for first element of this chunk: previous chunk's last eps,
        // or the carried value from the previous tile for lane 0.
        float eprev0 = (lane == 0) ? epsCarry : sE[lane * STRD - STRD + (KCH - 1)];

        // Local composition over the chunk: g_out = A*g_in + Bv
        float A = 1.0f, Bv = 0.0f;
        {
            float prev = eprev0;
#pragma unroll
            for (int j = 0; j < KCH; ++j) {
                float a = 1.0f - h2[j];
                float b = h2[j] * prev;
                Bv = fmaf(a, Bv, b);   // compose: a*(A g + Bv) + b
                A  = a * A;
                prev = e[j];
            }
        }

        // Inclusive Hillis-Steele scan across 32 lanes under composition
#pragma unroll
        for (int d = 1; d < 32; d <<= 1) {
            float Ap = __shfl_up(A, (unsigned)d, 32);
            float Bp = __shfl_up(Bv, (unsigned)d, 32);
            if (lane >= d) { Bv = fmaf(A, Bp, Bv); A = A * Ap; }
        }
        // Exclusive prefix -> gamma entering this lane's chunk
        float Aex = __shfl_up(A, 1u, 32);
        float Bex = __shfl_up(Bv, 1u, 32);
        if (lane == 0) { Aex = 1.0f; Bex = 0.0f; }
        float g = fmaf(Aex, gTile, Bex);

        // Carries for next tile (inclusive composition of whole tile = lane 31)
        float A31 = __shfl(A, 31, 32);
        float B31 = __shfl(Bv, 31, 32);
        float eL  = __shfl(e[KCH - 1], 31, 32);

        // Replay: sig_t = 2.5*e_t - 2*g (old g), then g <- g + 2h*(eps_{t-1} - g)
        v4f o[4];
        float* op = (float*)o;
        {
            float prev = eprev0;
#pragma unroll
            for (int j = 0; j < KCH; ++j) {
                op[j] = fmaf(2.5f, e[j], -(g + g));
                g = fmaf(h2[j], prev - g, g);
                prev = e[j];
            }
        }

        // Non-temporal stores: whole 512-elem tile region fully covered by the wave
        float* dst = rowO + tile * TILE_ELEMS + lane * KCH;
#pragma unroll
        for (int q = 0; q < 4; ++q)
            __builtin_nontemporal_store(o[q], (v4f*)dst + q);

        gTile    = fmaf(A31, gTile, B31);
        epsCarry = eL;
        buf = nb;
    }
}

extern "C" void kernel_launch(void* const* d_in, const int* in_sizes, int n_in,
                              void* d_out, int out_size, void* d_ws, size_t ws_size,
                              hipStream_t stream) {
    (void)in_sizes; (void)n_in; (void)d_ws; (void)ws_size; (void)out_size;
    const float* eps = (const float*)d_in[0];
    const float* hs  = (const float*)d_in[1];
    float* out       = (float*)d_out;

    dim3 grid(B_ROWS / WPB);   // 1024 blocks
    dim3 block(WPB * 32);      // 128 threads = 4 waves, one row per wave
    visco_scan_kernel<<<grid, block, 0, stream>>>(eps, hs, out);
}